// MPNN_90039694393773
// MI455X (gfx1250) — compile-verified
//
#include <hip/hip_runtime.h>
#include <hip/hip_bf16.h>

#define HID 64

typedef __attribute__((ext_vector_type(2))) float v2f;
typedef __attribute__((ext_vector_type(8))) float v8f;

// ---------------------------------------------------------------------------
// Encoder stage 1: t = relu(x @ enc_w1 + enc_b1), x:[n,4], w1:[4,64]
// ---------------------------------------------------------------------------
__global__ void enc1_kernel(const float* __restrict__ x,
                            const float* __restrict__ w1,
                            const float* __restrict__ b1,
                            float* __restrict__ t, int n) {
  int i = blockIdx.x * blockDim.x + threadIdx.x;
  if (i >= n) return;
  const float4 xv = *(const float4*)(x + (long long)i * 4);
#pragma unroll 8
  for (int j = 0; j < HID; ++j) {
    float acc = b1[j];
    acc += xv.x * w1[0 * HID + j];
    acc += xv.y * w1[1 * HID + j];
    acc += xv.z * w1[2 * HID + j];
    acc += xv.w * w1[3 * HID + j];
    t[(long long)i * HID + j] = fmaxf(acc, 0.0f);
  }
}

// ---------------------------------------------------------------------------
// WMMA GEMM: out[n,64] = A[n,64] @ W[64,64]  (+ optional bias, relu)
//
// Per block (256 thr = 8 waves):
//   1) async-DMA the 16KB weight matrix into LDS (global_load_async_to_lds_b128,
//      tracked by ASYNCcnt),
//   2) re-layout in LDS so the B-fragment pair (W[k][c], W[k+1][c]) is an
//      aligned float2 -> one ds_load_b64 per fragment,
//   3) each wave computes a 16x64 tile: 16 K-steps x 4 N-tiles of
//      v_wmma_f32_16x16x4_f32, A fragments streamed as global_load_b64.
// ---------------------------------------------------------------------------
__global__ void __launch_bounds__(256)
gemm_wmma_kernel(const float* __restrict__ A,
                 const float* __restrict__ W,
                 float* __restrict__ out,
                 const float* __restrict__ bias,
                 int n, int fuse_bias_relu) {
  __shared__ float rawW[HID * HID];   // row-major copy of W      (16KB)
  __shared__ float intW[HID * HID];   // K-pair interleaved copy  (16KB)

  const int tid = threadIdx.x;        // 0..255

  // --- stage 1: async copy W (4096 floats) global -> LDS, 16B per request ---
#pragma unroll
  for (int it = 0; it < 4; ++it) {
    int e4 = it * 256 + tid;                          // float4 index 0..1023
    unsigned lds = (unsigned)(unsigned long long)&rawW[e4 * 4];  // low 32b = LDS offset
    const float* g = W + e4 * 4;
    asm volatile("global_load_async_to_lds_b128 %0, %1, off"
                 :: "v"(lds), "v"(g) : "memory");
  }
  asm volatile("s_wait_asynccnt 0x0" ::: "memory");
  __syncthreads();

  // --- stage 2: interleave K pairs: intW[kk*128 + c*2 + p] = W[2kk+p][c] ---
#pragma unroll
  for (int it = 0; it < 8; ++it) {
    int p = it * 256 + tid;                           // pair index 0..2047
    int kk = p >> 6;                                  // 0..31
    int c  = p & 63;
    float2 pr = make_float2(rawW[(2 * kk + 0) * HID + c],
                            rawW[(2 * kk + 1) * HID + c]);
    *(float2*)&intW[kk * 128 + c * 2] = pr;
  }
  __syncthreads();

  // --- stage 3: WMMA ---
  const int lane = tid & 31;
  const int waveInBlock = tid >> 5;
  const long long wave = (long long)blockIdx.x * 8 + waveInBlock;
  const long long row0 = wave * 16;
  if (row0 < n) {                      // wave-uniform: EXEC stays all-1s
    const int lh = lane & 15;
    const int kh = lane >> 4;          // 0 or 1 (K half)

    long long r = row0 + lh;
    if (r >= n) r = n - 1;             // clamp A reads, keep EXEC full
    const float* __restrict__ arow = A + r * HID;

    v8f c0 = {}, c1 = {}, c2 = {}, c3 = {};
#pragma unroll
    for (int k = 0; k < HID; k += 4) {
      const float2 av = *(const float2*)(arow + k + 2 * kh);
      v2f a; a.x = av.x; a.y = av.y;

      const int kk = (k >> 1) + kh;    // interleaved pair row
      const float* fb = &intW[kk * 128];
      float2 f0 = *(const float2*)(fb + (0  + lh) * 2);
      float2 f1 = *(const float2*)(fb + (16 + lh) * 2);
      float2 f2 = *(const float2*)(fb + (32 + lh) * 2);
      float2 f3 = *(const float2*)(fb + (48 + lh) * 2);
      v2f b0, b1, b2, b3;
      b0.x = f0.x; b0.y = f0.y;
      b1.x = f1.x; b1.y = f1.y;
      b2.x = f2.x; b2.y = f2.y;
      b3.x = f3.x; b3.y = f3.y;

      c0 = __builtin_amdgcn_wmma_f32_16x16x4_f32(false, a, false, b0, (short)0, c0, false, false);
      c1 = __builtin_amdgcn_wmma_f32_16x16x4_f32(false, a, false, b1, (short)0, c1, false, false);
      c2 = __builtin_amdgcn_wmma_f32_16x16x4_f32(false, a, false, b2, (short)0, c2, false, false);
      c3 = __builtin_amdgcn_wmma_f32_16x16x4_f32(false, a, false, b3, (short)0, c3, false, false);
    }

    v8f cc[4] = {c0, c1, c2, c3};
#pragma unroll
    for (int nt = 0; nt < 4; ++nt) {
#pragma unroll
      for (int rr = 0; rr < 8; ++rr) {
        long long orow = row0 + rr + 8 * kh;
        int ocol = nt * 16 + lh;
        if (orow < n) {
          float v = cc[nt][rr];
          if (fuse_bias_relu) v = fmaxf(v + bias[ocol], 0.0f);
          out[orow * HID + ocol] = v;
        }
      }
    }
  }
}

// ---------------------------------------------------------------------------
// Graph prep
// ---------------------------------------------------------------------------
__global__ void init_st_kernel(int* st) { st[0] = 0x7FFFFFFF; st[1] = 0x7FFFFFFF; }

__global__ void prep_kernel(const float* __restrict__ x, float* __restrict__ deg,
                            int* st, int n) {
  int i = blockIdx.x * blockDim.x + threadIdx.x;
  if (i >= n) return;
  deg[i] = 1.0f;                                 // self-loop
  if (x[i * 4 + 2] == 1.0f) atomicMin(&st[0], i);
  if (x[i * 4 + 3] == 1.0f) atomicMin(&st[1], i);
}

__global__ void deg_kernel(const int* __restrict__ dst, float* __restrict__ deg, int nE) {
  int e = blockIdx.x * blockDim.x + threadIdx.x;
  if (e >= nE) return;
  atomicAdd(&deg[dst[e]], 1.0f);
}

__global__ void rsqrt_kernel(float* __restrict__ dis, int n) {
  int i = blockIdx.x * blockDim.x + threadIdx.x;
  if (i >= n) return;
  dis[i] = rsqrtf(dis[i]);                       // deg >= 1 always
}

// ---------------------------------------------------------------------------
// agg = dis[i]^2 * hw[i]   (self-loop term), float4 per thread
// ---------------------------------------------------------------------------
__global__ void agg_init_kernel(const float* __restrict__ hw,
                                const float* __restrict__ dis,
                                float* __restrict__ agg, int n) {
  long long idx = (long long)blockIdx.x * blockDim.x + threadIdx.x;
  if (idx >= (long long)n * 16) return;
  long long node = idx >> 4;
  int c = (int)(idx & 15) * 4;
  float s = dis[node]; s = s * s;
  long long base = node * HID + c;
  float4 v = *(const float4*)(hw + base);
  v.x *= s; v.y *= s; v.z *= s; v.w *= s;
  *(float4*)(agg + base) = v;
}

// ---------------------------------------------------------------------------
// Edge scatter: agg[dst] += dis[src]*dis[dst] * hw[src]; 16 threads/edge
// ---------------------------------------------------------------------------
__global__ void scatter_kernel(const float* __restrict__ hw,
                               const float* __restrict__ dis,
                               const int* __restrict__ src,
                               const int* __restrict__ dst,
                               float* __restrict__ agg, int nE) {
  long long idx = (long long)blockIdx.x * blockDim.x + threadIdx.x;
  if (idx >= (long long)nE * 16) return;
  long long e = idx >> 4;
  int c = (int)(idx & 15) * 4;
  int s = src[e], d = dst[e];
  float nrm = dis[s] * dis[d];
  const float4 hv = *(const float4*)(hw + (long long)s * HID + c);
  float* ad = agg + (long long)d * HID + c;
  atomicAdd(ad + 0, hv.x * nrm);
  atomicAdd(ad + 1, hv.y * nrm);
  atomicAdd(ad + 2, hv.z * nrm);
  atomicAdd(ad + 3, hv.w * nrm);
}

// ---------------------------------------------------------------------------
// h += relu(agg + b)
// ---------------------------------------------------------------------------
__global__ void update_kernel(float* __restrict__ h, const float* __restrict__ agg,
                              const float* __restrict__ b, int n) {
  long long idx = (long long)blockIdx.x * blockDim.x + threadIdx.x;
  if (idx >= (long long)n * 16) return;
  long long node = idx >> 4;
  int c = (int)(idx & 15) * 4;
  long long base = node * HID + c;
  float4 hv = *(const float4*)(h + base);
  const float4 av = *(const float4*)(agg + base);
  const float4 bv = *(const float4*)(b + c);
  hv.x += fmaxf(av.x + bv.x, 0.0f);
  hv.y += fmaxf(av.y + bv.y, 0.0f);
  hv.z += fmaxf(av.z + bv.z, 0.0f);
  hv.w += fmaxf(av.w + bv.w, 0.0f);
  *(float4*)(h + base) = hv;
}

// ---------------------------------------------------------------------------
// Head MLP, single block of 128 threads
// ---------------------------------------------------------------------------
__global__ void head_kernel(const float* __restrict__ h, const int* __restrict__ st,
                            const float* __restrict__ w1, const float* __restrict__ b1,
                            const float* __restrict__ w2, const float* __restrict__ b2,
                            const float* __restrict__ w3, const float* __restrict__ b3,
                            float* __restrict__ out, int n) {
  __shared__ float comb[2 * HID];
  __shared__ float o1[HID];
  __shared__ float o2[HID / 2];
  int s = st[0], t = st[1];
  if (s == 0x7FFFFFFF || t == 0x7FFFFFFF) { s = 0; t = n - 1; }   // "ok" fallback
  int tid = threadIdx.x;
  if (tid < HID) comb[tid] = h[(long long)s * HID + tid];
  else           comb[tid] = h[(long long)t * HID + (tid - HID)];
  __syncthreads();
  if (tid < HID) {
    float acc = b1[tid];
    for (int k = 0; k < 2 * HID; ++k) acc += comb[k] * w1[k * HID + tid];
    o1[tid] = fmaxf(acc, 0.0f);
  }
  __syncthreads();
  if (tid < HID / 2) {
    float acc = b2[tid];
    for (int k = 0; k < HID; ++k) acc += o1[k] * w2[k * (HID / 2) + tid];
    o2[tid] = fmaxf(acc, 0.0f);
  }
  __syncthreads();
  if (tid == 0) {
    float acc = b3[0];
    for (int k = 0; k < HID / 2; ++k) acc += o2[k] * w3[k];
    out[0] = acc;
  }
}

// ---------------------------------------------------------------------------
extern "C" void kernel_launch(void* const* d_in, const int* in_sizes, int n_in,
                              void* d_out, int out_size, void* d_ws, size_t ws_size,
                              hipStream_t stream) {
  const float* x      = (const float*)d_in[0];
  const int*   ei     = (const int*)d_in[1];     // [2, E] flat (int32)
  // d_in[2] = edge_attr (unused by the reference math)
  const float* enc_w1 = (const float*)d_in[3];
  const float* enc_b1 = (const float*)d_in[4];
  const float* enc_w2 = (const float*)d_in[5];
  const float* enc_b2 = (const float*)d_in[6];
  const float* conv_w = (const float*)d_in[7];   // [3,64,64]
  const float* conv_b = (const float*)d_in[8];   // [3,64]
  const float* out_w1 = (const float*)d_in[9];
  const float* out_b1 = (const float*)d_in[10];
  const float* out_w2 = (const float*)d_in[11];
  const float* out_b2 = (const float*)d_in[12];
  const float* out_w3 = (const float*)d_in[13];
  const float* out_b3 = (const float*)d_in[14];
  float* out = (float*)d_out;

  const int n  = in_sizes[0] / 4;        // NODE_F = 4
  const int nE = in_sizes[2];            // edge_attr: [E,1]
  const int* src = ei;
  const int* dst = ei + nE;

  // workspace carve-out (~77 MB, fully L2-resident on MI455X)
  float* ws  = (float*)d_ws;
  float* h   = ws;                               // n*64
  float* hw  = h  + (long long)n * HID;          // n*64
  float* agg = hw + (long long)n * HID;          // n*64
  float* dis = agg + (long long)n * HID;         // n
  int*   st  = (int*)(dis + n);                  // 2

  const int B = 256;
  const int gN   = (n + B - 1) / B;
  const int gE   = (nE + B - 1) / B;
  const int gN16 = (int)(((long long)n * 16 + B - 1) / B);
  const int gE16 = (int)(((long long)nE * 16 + B - 1) / B);
  const int nWaves = (n + 15) / 16;
  const int gGemm  = (nWaves + (B / 32) - 1) / (B / 32);

  // encoder
  enc1_kernel<<<gN, B, 0, stream>>>(x, enc_w1, enc_b1, hw, n);
  gemm_wmma_kernel<<<gGemm, B, 0, stream>>>(hw, enc_w2, h, enc_b2, n, 1);

  // gcn_norm with self-loops + source/target discovery
  init_st_kernel<<<1, 1, 0, stream>>>(st);
  prep_kernel<<<gN, B, 0, stream>>>(x, dis, st, n);
  deg_kernel<<<gE, B, 0, stream>>>(dst, dis, nE);
  rsqrt_kernel<<<gN, B, 0, stream>>>(dis, n);

  // 3 conv layers
  for (int l = 0; l < 3; ++l) {
    const float* Wl = conv_w + (long long)l * HID * HID;
    const float* bl = conv_b + (long long)l * HID;
    gemm_wmma_kernel<<<gGemm, B, 0, stream>>>(h, Wl, hw, nullptr, n, 0);
    agg_init_kernel<<<gN16, B, 0, stream>>>(hw, dis, agg, n);
    scatter_kernel<<<gE16, B, 0, stream>>>(hw, dis, src, dst, agg, nE);
    update_kernel<<<gN16, B, 0, stream>>>(h, agg, bl, n);
  }

  // head
  head_kernel<<<1, 128, 0, stream>>>(h, st, out_w1, out_b1, out_w2, out_b2,
                                     out_w3, out_b3, out, n);
}